// AGNN_73478300500623
// MI455X (gfx1250) — compile-verified
//
#include <hip/hip_runtime.h>
#include <hip/hip_bf16.h>
#include <math.h>

// ---------------------------------------------------------------------------
// AGNN: h = relu(x @ W1^T + b1); h = AGNNConv(h, beta=1); h = AGNNConv(h, beta2)
// N=50000 nodes, 600000 edges (+50000 self loops), IN=256, HID=128, fp32.
// GEMM via V_WMMA_F32_16X16X4_F32 (fp32-exact; GEMM is memory-bound so the
// fp32 WMMA rate is irrelevant). Inner loop batches 18 independent loads so
// the scheduler can overlap them instead of load->wait(0)->wmma serialization.
// Conv passes: wave-per-edge gather/scatter (512B per node row = ideal HBM
// granularity), segment softmax via monotonic-uint atomicMax + f32 atomicAdd.
// ---------------------------------------------------------------------------

typedef float v2f __attribute__((ext_vector_type(2)));
typedef float v8f __attribute__((ext_vector_type(8)));

#define N_NODES 50000
#define N_EDGES 600000
#define E_TOT   650000   // edges + self loops
#define IN_CH   256
#define HID     128
#define ENC_NEG_INF 0x007FFFFFu  // enc_f32(-inf)

// monotonic float <-> uint mapping for atomicMax on signed floats
__device__ __forceinline__ unsigned enc_f32(float f) {
  unsigned u = __float_as_uint(f);
  return (u & 0x80000000u) ? ~u : (u | 0x80000000u);
}
__device__ __forceinline__ float dec_f32(unsigned e) {
  unsigned u = (e & 0x80000000u) ? (e & 0x7FFFFFFFu) : ~e;
  return __uint_as_float(u);
}

__device__ __forceinline__ float wave_sum(float v) {
  // wave32 butterfly reduction
  #pragma unroll
  for (int off = 16; off >= 1; off >>= 1) v += __shfl_xor(v, off, 32);
  return v;
}

// ---------------------------------------------------------------------------
// GEMM + bias + ReLU + fused row-rnorm:
//   H[50000,128] = relu(X[50000,256] @ W^T + b);  rn[i] = 1/max(||H_i||,1e-12)
// One wave per 16-row tile. Per double-k-step: 2 A frags + 16 B frags loaded
// up-front (18 loads in flight), then 16 WMMAs.
// A layout (ISA 16x4 f32): lane l -> row (l&15), K = (l>>4)*2 + {0,1}
// B layout (4x16 f32):     lane l -> col (l&15), K = (l>>4)*2 + {0,1}
// C/D layout: VGPR r, lane l -> row r + (l>>4)*8, col (l&15)
// ---------------------------------------------------------------------------
__global__ void k_gemm_relu(const float* __restrict__ X,
                            const float* __restrict__ W,
                            const float* __restrict__ B,
                            float* __restrict__ H,
                            float* __restrict__ rn) {
  const int wave = (int)((blockIdx.x * blockDim.x + threadIdx.x) >> 5);
  const int lane = (int)(threadIdx.x & 31);
  if (wave * 16 >= N_NODES) return;  // whole-wave guard: EXEC stays all-ones

  const int m0   = wave * 16;
  const int mrow = m0 + (lane & 15);
  const int kg   = (lane >> 4) * 2;   // 0 or 2
  const int n15  = lane & 15;

  v8f acc[8];
  #pragma unroll
  for (int t = 0; t < 8; ++t) acc[t] = (v8f){0.f,0.f,0.f,0.f,0.f,0.f,0.f,0.f};

  const float* __restrict__ xrow = X + (size_t)mrow * IN_CH + kg;
  const float* __restrict__ wbas = W + (size_t)n15 * IN_CH + kg;

  for (int k = 0; k < IN_CH; k += 8) {
    // ---- issue all 18 independent loads first (deep MLP, L2-resident W) ----
    const v2f a0 = *(const v2f*)(xrow + k);
    const v2f a1 = *(const v2f*)(xrow + k + 4);
    v2f b0[8], b1[8];
    #pragma unroll
    for (int t = 0; t < 8; ++t)
      b0[t] = *(const v2f*)(wbas + (size_t)t * 16 * IN_CH + k);
    #pragma unroll
    for (int t = 0; t < 8; ++t)
      b1[t] = *(const v2f*)(wbas + (size_t)t * 16 * IN_CH + k + 4);
    // ---- then the WMMA chain ----
    #pragma unroll
    for (int t = 0; t < 8; ++t)
      acc[t] = __builtin_amdgcn_wmma_f32_16x16x4_f32(
          false, a0, false, b0[t], (short)0, acc[t], false, false);
    #pragma unroll
    for (int t = 0; t < 8; ++t)
      acc[t] = __builtin_amdgcn_wmma_f32_16x16x4_f32(
          false, a1, false, b1[t], (short)0, acc[t], false, false);
  }

  // Epilogue: bias + relu + store, accumulating per-row sum of squares.
  const int mside = (lane >> 4) * 8;
  float ss[8];
  #pragma unroll
  for (int r = 0; r < 8; ++r) ss[r] = 0.f;

  #pragma unroll
  for (int t = 0; t < 8; ++t) {
    const int col = t * 16 + n15;
    const float bb = B[col];
    #pragma unroll
    for (int r = 0; r < 8; ++r) {
      const int row = m0 + r + mside;
      float v = acc[t][r] + bb;
      v = v > 0.f ? v : 0.f;
      H[(size_t)row * HID + col] = v;
      ss[r] += v * v;
    }
  }
  // Reduce Σv² across the 16 lanes of each half-wave (rows are disjoint per
  // half: lanes 0-15 hold rows m0+r, lanes 16-31 hold rows m0+8+r).
  #pragma unroll
  for (int r = 0; r < 8; ++r) {
    #pragma unroll
    for (int off = 8; off >= 1; off >>= 1) ss[r] += __shfl_xor(ss[r], off, 32);
  }
  if (n15 == 0) {
    #pragma unroll
    for (int r = 0; r < 8; ++r)
      rn[m0 + r + mside] = 1.f / fmaxf(sqrtf(ss[r]), 1e-12f);
  }
}

// ---------------------------------------------------------------------------
// Per-conv kernels
// ---------------------------------------------------------------------------
__global__ void k_init(float* __restrict__ Y,
                       unsigned* __restrict__ segmax,
                       float* __restrict__ segsum) {
  const int i = (int)(blockIdx.x * blockDim.x + threadIdx.x);
  if (i < N_NODES * HID) Y[i] = 0.f;
  if (i < N_NODES) { segmax[i] = ENC_NEG_INF; segsum[i] = 0.f; }
}

// 1/max(||x_i||, 1e-12), wave per node (lane holds float4 = 4 channels)
__global__ void k_rnorm(const float* __restrict__ X, float* __restrict__ rn) {
  const int node = (int)((blockIdx.x * blockDim.x + threadIdx.x) >> 5);
  const int lane = (int)(threadIdx.x & 31);
  if (node >= N_NODES) return;
  const float4 v = *(const float4*)(X + (size_t)node * HID + lane * 4);
  float s = v.x * v.x + v.y * v.y + v.z * v.z + v.w * v.w;
  s = wave_sum(s);
  if (lane == 0) rn[node] = 1.f / fmaxf(sqrtf(s), 1e-12f);
}

// logit_e = beta * cos(x_src, x_dst); segment max over dst via atomicMax
__global__ void k_edge_logit(const float* __restrict__ X,
                             const int* __restrict__ EI,
                             const float* __restrict__ rn,
                             const float* __restrict__ beta_ptr,
                             float* __restrict__ elog,
                             unsigned* __restrict__ segmax) {
  const int e    = (int)((blockIdx.x * blockDim.x + threadIdx.x) >> 5);
  const int lane = (int)(threadIdx.x & 31);
  if (e >= E_TOT) return;
  int src, dst;
  if (e < N_EDGES) { src = EI[e]; dst = EI[N_EDGES + e]; }
  else             { src = dst = e - N_EDGES; }  // self loop
  const float4 a = *(const float4*)(X + (size_t)src * HID + lane * 4);
  const float4 b = *(const float4*)(X + (size_t)dst * HID + lane * 4);
  float s = a.x * b.x + a.y * b.y + a.z * b.z + a.w * b.w;
  s = wave_sum(s);
  if (lane == 0) {
    const float beta = beta_ptr ? beta_ptr[0] : 1.0f;
    const float logit = beta * s * rn[src] * rn[dst];
    elog[e] = logit;
    atomicMax(segmax + dst, enc_f32(logit));
  }
}

// e = exp(logit - segmax[dst]); segment sum over dst
__global__ void k_edge_exp(const int* __restrict__ EI,
                           const unsigned* __restrict__ segmax,
                           float* __restrict__ elog,
                           float* __restrict__ segsum) {
  const int e = (int)(blockIdx.x * blockDim.x + threadIdx.x);
  if (e >= E_TOT) return;
  const int dst = (e < N_EDGES) ? EI[N_EDGES + e] : (e - N_EDGES);
  const float ex = __expf(elog[e] - dec_f32(segmax[dst]));
  elog[e] = ex;
  atomicAdd(segsum + dst, ex);
}

// Y[dst] += alpha_e * X[src], wave per edge, 4 channels per lane
__global__ void k_scatter(const float* __restrict__ X,
                          const int* __restrict__ EI,
                          const float* __restrict__ elog,
                          const float* __restrict__ segsum,
                          float* __restrict__ Y) {
  const int e    = (int)((blockIdx.x * blockDim.x + threadIdx.x) >> 5);
  const int lane = (int)(threadIdx.x & 31);
  if (e >= E_TOT) return;
  int src, dst;
  if (e < N_EDGES) { src = EI[e]; dst = EI[N_EDGES + e]; }
  else             { src = dst = e - N_EDGES; }
  const float alpha = elog[e] / (segsum[dst] + 1e-16f);
  const float4 xs = *(const float4*)(X + (size_t)src * HID + lane * 4);
  float* yp = Y + (size_t)dst * HID + lane * 4;
  atomicAdd(yp + 0, alpha * xs.x);
  atomicAdd(yp + 1, alpha * xs.y);
  atomicAdd(yp + 2, alpha * xs.z);
  atomicAdd(yp + 3, alpha * xs.w);
}

// ---------------------------------------------------------------------------
static void run_conv(const float* X, float* Y, const int* EI,
                     const float* beta_ptr, float* rn, unsigned* segmax,
                     float* segsum, float* elog, bool compute_rnorm,
                     hipStream_t stream) {
  const int init_threads = N_NODES * HID;                       // 6.4M
  k_init<<<(init_threads + 255) / 256, 256, 0, stream>>>(Y, segmax, segsum);

  if (compute_rnorm) {
    const long long norm_threads = (long long)N_NODES * 32;     // wave/node
    k_rnorm<<<(int)((norm_threads + 255) / 256), 256, 0, stream>>>(X, rn);
  }

  const long long edge_threads = (long long)E_TOT * 32;         // wave/edge
  const int eblocks = (int)((edge_threads + 255) / 256);        // 81250
  k_edge_logit<<<eblocks, 256, 0, stream>>>(X, EI, rn, beta_ptr, elog, segmax);

  k_edge_exp<<<(E_TOT + 255) / 256, 256, 0, stream>>>(EI, segmax, elog, segsum);

  k_scatter<<<eblocks, 256, 0, stream>>>(X, EI, elog, segsum, Y);
}

extern "C" void kernel_launch(void* const* d_in, const int* in_sizes, int n_in,
                              void* d_out, int out_size, void* d_ws, size_t ws_size,
                              hipStream_t stream) {
  const float* x     = (const float*)d_in[0];   // [50000, 256]
  const int*   ei    = (const int*)d_in[1];     // [2, 600000]
  const float* W1    = (const float*)d_in[2];   // [128, 256]
  const float* b1    = (const float*)d_in[3];   // [128]
  const float* beta2 = (const float*)d_in[4];   // [1]
  float* out = (float*)d_out;                   // [50000, 128]

  float*    ws     = (float*)d_ws;
  float*    h1     = ws;                        // 6,400,000 f
  float*    h2     = ws + 6400000;              // 6,400,000 f
  float*    rn     = ws + 12800000;             //    50,000 f
  unsigned* segmax = (unsigned*)(ws + 12850000);//    50,000 u32
  float*    segsum = ws + 12900000;             //    50,000 f
  float*    elog   = ws + 12950000;             //   650,000 f
  // total: 13,600,000 floats = 54.4 MB of workspace

  // GEMM: 3125 16-row tiles, 8 waves per 256-thread block -> 391 blocks.
  // Also produces rn for prop1 (fused epilogue) -> no k_rnorm pass for conv1.
  const int gemm_waves  = (N_NODES + 15) / 16;                  // 3125
  const int gemm_blocks = (gemm_waves * 32 + 255) / 256;        // 391
  k_gemm_relu<<<gemm_blocks, 256, 0, stream>>>(x, W1, b1, h1, rn);

  // prop1: fixed beta = 1.0 (rnorm already computed by GEMM epilogue)
  run_conv(h1, h2, ei, nullptr, rn, segmax, segsum, elog, false, stream);
  // prop2: learnable beta (device scalar), rnorm of h2 computed here
  run_conv(h2, out, ei, beta2, rn, segmax, segsum, elog, true, stream);
}